// BrownianBridgeCriterion_21337397526846
// MI455X (gfx1250) — compile-verified
//
#include <hip/hip_runtime.h>
#include <hip/hip_bf16.h>

#define BS_ 16
#define T_  16
#define Q_  100
#define HID_ 256
#define PROJ_ 256
#define NROWS_ 1600           // BS*Q
#define NNEG_  3200           // 2*NROWS
#define MTOT_  25600          // BS*T*Q

typedef float v2f __attribute__((ext_vector_type(2)));
typedef float v8f __attribute__((ext_vector_type(8)));

// ---- gfx1250 async global->LDS path (guarded; falls back to reg staging) ----
#if defined(__has_builtin)
#if __has_builtin(__builtin_amdgcn_global_load_async_to_lds_b32)
#define HAVE_ASYNC_LDS 1
#endif
#if __has_builtin(__builtin_amdgcn_s_wait_asynccnt)
#define HAVE_ASYNC_WAIT_BUILTIN 1
#endif
#endif
#ifndef HAVE_ASYNC_LDS
#define HAVE_ASYNC_LDS 0
#endif

#if HAVE_ASYNC_LDS
typedef __attribute__((address_space(1))) int* as1_i32p;
typedef __attribute__((address_space(1))) void* as1_vp;
typedef __attribute__((address_space(3))) int* as3_i32p;
typedef __attribute__((address_space(3))) void* as3_vp;
__device__ __forceinline__ void async_g2l_b32(const float* gsrc, float* ldst) {
    __builtin_amdgcn_global_load_async_to_lds_b32(
        (as1_i32p)(as1_vp)(void*)(float*)gsrc,
        (as3_i32p)(as3_vp)(void*)ldst, 0, 0);
}
#ifdef HAVE_ASYNC_WAIT_BUILTIN
#define ASYNC_WAIT0() __builtin_amdgcn_s_wait_asynccnt(0)
#else
#define ASYNC_WAIT0() asm volatile("s_wait_asynccnt 0" ::: "memory")
#endif
#endif

// ---------------------------------------------------------------------------
// Workspace layout (float units)
// ---------------------------------------------------------------------------
#define OFF_X      0
#define OFF_A      (OFF_X + (size_t)NNEG_ * T_ * PROJ_)
#define OFF_NUMER  (OFF_A + (size_t)NROWS_ * PROJ_)
#define OFF_SIG2   (OFF_NUMER + NROWS_)
#define OFF_AA     (OFF_SIG2 + NROWS_)
#define OFF_LOSS   (OFF_AA + NROWS_)
#define OFF_HT     (OFF_LOSS + NROWS_)
#define OFF_INTS   (OFF_HT + NROWS_)
#define OFF_CROSS  (OFF_INTS + 4096)
// int sub-offsets inside ints region
#define I_COUNT  0
#define I_OFFS   16
#define I_CURSOR 32
#define I_GROUPT 48    // 128 entries
#define I_PERM   176   // 2048 entries
#define I_TOTAL  2224

__device__ __forceinline__ float block_reduce_sum_256(float v, float* sh) {
    int tid = threadIdx.x;
    sh[tid] = v;
    __syncthreads();
    for (int s = 128; s > 0; s >>= 1) {
        if (tid < s) sh[tid] += sh[tid + s];
        __syncthreads();
    }
    float r = sh[0];
    __syncthreads();
    return r;
}

// ---------------------------------------------------------------------------
__global__ void k_init(int* ints) {
    for (int s = threadIdx.x; s < I_TOTAL; s += 256)
        ints[s] = (s < I_GROUPT) ? 0 : -1;
}

// ---------------------------------------------------------------------------
// Projection GEMM: x = src @ W + b, scatter (b,t,q,:) -> (n=b*Q+q, t, :)
// Block = 256 threads (8 waves). Each block: 128 M rows x 64 N cols.
// W tile (256K x 64N) staged ONCE per block into LDS, transposed to [n][k]
// (+4 pad) via async global->LDS, so B fragments are contiguous ds_load_b64.
// grid = (MTOT/128, PROJ/64), block = 256
// ---------------------------------------------------------------------------
#define LDS_KP 260   // 256 + 4 pad: lane n-stride = 260 dwords -> distinct banks
__global__ void k_proj(const float* __restrict__ src, const float* __restrict__ Wm,
                       const float* __restrict__ bias, float* __restrict__ dst) {
    __shared__ float smemT[64 * LDS_KP];   // ~66.6 KB of 320 KB/WGP

    const int tid   = threadIdx.x;
    const int nbase = blockIdx.y * 64;

    // ---- stage W tile transposed: smemT[n][k] = W[k][nbase+n] ----
    {
        const int nloc  = tid & 63;        // lanes consecutive in n -> coalesced 128B
        const int kbase = tid >> 6;        // 0..3
#pragma unroll 8
        for (int i = 0; i < 64; ++i) {
            const int k = i * 4 + kbase;
#if HAVE_ASYNC_LDS
            async_g2l_b32(Wm + (size_t)k * PROJ_ + nbase + nloc,
                          &smemT[nloc * LDS_KP + k]);
#else
            smemT[nloc * LDS_KP + k] = Wm[(size_t)k * PROJ_ + nbase + nloc];
#endif
        }
#if HAVE_ASYNC_LDS
        ASYNC_WAIT0();
#endif
    }
    __syncthreads();

    const int lane  = tid & 31;
    const int wv    = tid >> 5;            // 0..7 : wave's M sub-tile
    const int il    = lane & 15;
    const int half  = lane >> 4;
    const int khi   = half * 2;            // K+0/K+1 vs K+2/K+3
    const int mbase = blockIdx.x * 128 + wv * 16;
    const int mrow  = mbase + il;

    v8f acc0 = {}, acc1 = {}, acc2 = {}, acc3 = {};
    const float* b0p = &smemT[(0 + il)  * LDS_KP];
    const float* b1p = &smemT[(16 + il) * LDS_KP];
    const float* b2p = &smemT[(32 + il) * LDS_KP];
    const float* b3p = &smemT[(48 + il) * LDS_KP];
    for (int k0 = 0; k0 < HID_; k0 += 4) {
        const int kk = k0 + khi;
        v2f a;
        a.x = src[(size_t)mrow * HID_ + kk];
        a.y = src[(size_t)mrow * HID_ + kk + 1];
        v2f b0, b1, b2, b3;                 // contiguous in k -> ds_load_b64
        b0.x = b0p[kk]; b0.y = b0p[kk + 1];
        b1.x = b1p[kk]; b1.y = b1p[kk + 1];
        b2.x = b2p[kk]; b2.y = b2p[kk + 1];
        b3.x = b3p[kk]; b3.y = b3p[kk + 1];
        acc0 = __builtin_amdgcn_wmma_f32_16x16x4_f32(false, a, false, b0, (short)0, acc0, false, false);
        acc1 = __builtin_amdgcn_wmma_f32_16x16x4_f32(false, a, false, b1, (short)0, acc1, false, false);
        acc2 = __builtin_amdgcn_wmma_f32_16x16x4_f32(false, a, false, b2, (short)0, acc2, false, false);
        acc3 = __builtin_amdgcn_wmma_f32_16x16x4_f32(false, a, false, b3, (short)0, acc3, false, false);
    }
    // C/D layout: VGPR i -> (lanes 0-15: M=i, N=lane) (lanes 16-31: M=i+8, N=lane-16)
    const float bv0 = bias[nbase + il];
    const float bv1 = bias[nbase + 16 + il];
    const float bv2 = bias[nbase + 32 + il];
    const float bv3 = bias[nbase + 48 + il];
#pragma unroll
    for (int i = 0; i < 8; ++i) {
        int m   = mbase + i + half * 8;
        int bq  = m / (T_ * Q_);
        int rem = m % (T_ * Q_);
        int t   = rem / Q_;
        int q   = rem % Q_;
        size_t outrow = ((size_t)(bq * Q_ + q) * T_ + t) * PROJ_;
        dst[outrow + nbase + il]      = acc0[i] + bv0;
        dst[outrow + nbase + 16 + il] = acc1[i] + bv1;
        dst[outrow + nbase + 32 + il] = acc2[i] + bv2;
        dst[outrow + nbase + 48 + il] = acc3[i] + bv3;
    }
}

// ---------------------------------------------------------------------------
// Row L2 normalize: one block per row of 256
// ---------------------------------------------------------------------------
__global__ void k_norm(float* __restrict__ X) {
    __shared__ float sh[256];
    size_t row = blockIdx.x;
    float v = X[row * PROJ_ + threadIdx.x];
    float s = block_reduce_sum_256(v * v, sh);
    X[row * PROJ_ + threadIdx.x] = v * rsqrtf(s);
}

// ---------------------------------------------------------------------------
// Bridge prep: a[i], aa[i], sigma2[i], numer[i]; histogram bp
// ---------------------------------------------------------------------------
__global__ void k_bridge(const float* __restrict__ cur, const int* __restrict__ bridge,
                         float* __restrict__ A, float* __restrict__ numer,
                         float* __restrict__ sig2, float* __restrict__ aa,
                         int* __restrict__ counts) {
    __shared__ float sh[256];
    const int i = blockIdx.x;
    const int c = threadIdx.x;
    const int bh = bridge[i * 3 + 0];
    const int bp = bridge[i * 3 + 1];
    const int bt = bridge[i * 3 + 2];
    const float alpha = (float)(bp - bh) / (float)(bt - bh);
    const float sigma = alpha * (float)(bt - bp);

    const float g0 = cur[((size_t)i * T_ + bh) * PROJ_ + c];
    const float gp = cur[((size_t)i * T_ + bp) * PROJ_ + c];
    const float g2 = cur[((size_t)i * T_ + bt) * PROJ_ + c];
    const float ac = (1.0f - alpha) * g0 + alpha * g2;
    const float xc = gp - ac;
    A[(size_t)i * PROJ_ + c] = ac;
    float sx = block_reduce_sum_256(xc * xc, sh);
    float sa = block_reduce_sum_256(ac * ac, sh);
    if (c == 0) {
        float s2 = sigma * sigma;
        sig2[i]  = s2;
        aa[i]    = sa;
        numer[i] = __expf(-sx / (2.0f * s2));
        atomicAdd(&counts[bp & 15], 1);
    }
}

// ---------------------------------------------------------------------------
__global__ void k_buckets(const int* __restrict__ counts, int* __restrict__ offs,
                          int* __restrict__ group_t) {
    if (threadIdx.x == 0 && blockIdx.x == 0) {
        int off = 0;
        for (int t = 0; t < 16; ++t) {
            offs[t] = off;
            int ng = (counts[t] + 15) >> 4;
            for (int k = 0; k < ng; ++k) group_t[(off >> 4) + k] = t;
            off += ng << 4;
        }
    }
}

__global__ void k_scatter(const int* __restrict__ bridge, const int* __restrict__ offs,
                          int* __restrict__ cursor, int* __restrict__ perm) {
    int i = blockIdx.x * 256 + threadIdx.x;
    if (i < NROWS_) {
        int bp = bridge[i * 3 + 1] & 15;
        int pos = offs[bp] + atomicAdd(&cursor[bp], 1);
        perm[pos] = i;
    }
}

// ---------------------------------------------------------------------------
// Gathered cross GEMM: cross[i,j] = a[i] . neg_all[j, t_i, :]
// Both operands contiguous along K -> b64 global loads; no LDS needed
// (zero intra-block reuse of B; cross-block reuse served by 192MB L2).
// grid = (128 groups, 50 j-quads), block = 32; early-exit unused groups.
// ---------------------------------------------------------------------------
__global__ void k_cross(const float* __restrict__ A, const float* __restrict__ X,
                        const int* __restrict__ group_t, const int* __restrict__ perm,
                        float* __restrict__ cross) {
    const int g = blockIdx.x;
    const int t = group_t[g];
    if (t < 0) return;                    // uniform per-wave exit
    const int lane = threadIdx.x;
    const int il   = lane & 15;
    const int half = lane >> 4;
    const int khi  = half * 2;
    const int jb   = blockIdx.y * 64;
    const int idx  = perm[g * 16 + il];   // row for this lane's M slot (-1 = pad)

    v8f acc0 = {}, acc1 = {}, acc2 = {}, acc3 = {};
    const float* arow = (idx >= 0) ? (A + (size_t)idx * PROJ_) : nullptr;
    const float* b0b = X + ((size_t)(jb + il)      * T_ + t) * PROJ_;
    const float* b1b = X + ((size_t)(jb + 16 + il) * T_ + t) * PROJ_;
    const float* b2b = X + ((size_t)(jb + 32 + il) * T_ + t) * PROJ_;
    const float* b3b = X + ((size_t)(jb + 48 + il) * T_ + t) * PROJ_;
    for (int k0 = 0; k0 < PROJ_; k0 += 4) {
        const int kk = k0 + khi;
        v2f a;
        if (idx >= 0) { a.x = arow[kk]; a.y = arow[kk + 1]; }
        else          { a.x = 0.0f;     a.y = 0.0f; }
        v2f b0, b1, b2, b3;
        b0.x = b0b[kk]; b0.y = b0b[kk + 1];
        b1.x = b1b[kk]; b1.y = b1b[kk + 1];
        b2.x = b2b[kk]; b2.y = b2b[kk + 1];
        b3.x = b3b[kk]; b3.y = b3b[kk + 1];
        acc0 = __builtin_amdgcn_wmma_f32_16x16x4_f32(false, a, false, b0, (short)0, acc0, false, false);
        acc1 = __builtin_amdgcn_wmma_f32_16x16x4_f32(false, a, false, b1, (short)0, acc1, false, false);
        acc2 = __builtin_amdgcn_wmma_f32_16x16x4_f32(false, a, false, b2, (short)0, acc2, false, false);
        acc3 = __builtin_amdgcn_wmma_f32_16x16x4_f32(false, a, false, b3, (short)0, acc3, false, false);
    }
#pragma unroll
    for (int i = 0; i < 8; ++i) {
        int r  = i + half * 8;
        int ii = perm[g * 16 + r];
        if (ii >= 0) {
            float* crow = cross + (size_t)ii * NNEG_ + jb + il;
            crow[0]  = acc0[i];
            crow[16] = acc1[i];
            crow[32] = acc2[i];
            crow[48] = acc3[i];
        }
    }
}

// ---------------------------------------------------------------------------
// Per-row self + top-5 -> numer/deno ;  dist monotone in cross value
// ---------------------------------------------------------------------------
__global__ void k_topk(const float* __restrict__ cross, const float* __restrict__ numer,
                       const float* __restrict__ sig2, const float* __restrict__ aa,
                       float* __restrict__ lossbuf) {
    __shared__ float buf[256 * 5];
    __shared__ float selfv_sh;
    const int i   = blockIdx.x;
    const int tid = threadIdx.x;
    const float* row = cross + (size_t)i * NNEG_;

    float top[5] = {-1e30f, -1e30f, -1e30f, -1e30f, -1e30f};
    for (int j = tid; j < NNEG_; j += 256) {
        float v = row[j];
        if (j == i) { selfv_sh = v; continue; }
        if (v > top[4]) {
            top[4] = v;
#pragma unroll
            for (int k = 4; k > 0; --k)
                if (top[k] > top[k - 1]) { float tmp = top[k - 1]; top[k - 1] = top[k]; top[k] = tmp; }
        }
    }
#pragma unroll
    for (int k = 0; k < 5; ++k) buf[tid * 5 + k] = top[k];
    __syncthreads();
    if (tid == 0) {
        float best[5] = {-1e30f, -1e30f, -1e30f, -1e30f, -1e30f};
        for (int s = 0; s < 256 * 5; ++s) {
            float v = buf[s];
            if (v > best[4]) {
                best[4] = v;
                for (int k = 4; k > 0; --k)
                    if (best[k] > best[k - 1]) { float tmp = best[k - 1]; best[k - 1] = best[k]; best[k] = tmp; }
            }
        }
        const float av  = aa[i];
        const float inv = 1.0f / (2.0f * sig2[i]);
        // dist(c) = -(1 - 2c + ||a||^2) / (2 sigma^2)   (rows are unit-norm)
        float deno = __expf(-(1.0f - 2.0f * selfv_sh + av) * inv);
#pragma unroll
        for (int k = 0; k < 5; ++k)
            deno += __expf(-(1.0f - 2.0f * best[k] + av) * inv);
        lossbuf[i] = numer[i] / deno;
    }
}

// ---------------------------------------------------------------------------
__global__ void k_headtail(const float* __restrict__ cur, float* __restrict__ htbuf) {
    __shared__ float sh[256];
    const int n = blockIdx.x;
    float p = cur[((size_t)n * T_ + 0) * PROJ_ + threadIdx.x] *
              cur[((size_t)n * T_ + (T_ - 1)) * PROJ_ + threadIdx.x];
    float s = block_reduce_sum_256(p, sh);
    if (threadIdx.x == 0) htbuf[n] = log1pf(__expf(0.3f - s));
}

// ---------------------------------------------------------------------------
__global__ void k_final(const float* __restrict__ lossbuf, const float* __restrict__ htbuf,
                        float* __restrict__ out) {
    __shared__ float sh[256];
    const int tid = threadIdx.x;
    float s0 = 0.0f, s1 = 0.0f;
    for (int j = tid; j < NROWS_; j += 256) { s0 += lossbuf[j]; s1 += htbuf[j]; }
    float t0 = block_reduce_sum_256(s0, sh);
    float t1 = block_reduce_sum_256(s1, sh);
    if (tid == 0) {
        out[0] = t0 / (float)NROWS_;
        out[1] = t1 / (float)NROWS_;
    }
}

// ---------------------------------------------------------------------------
extern "C" void kernel_launch(void* const* d_in, const int* in_sizes, int n_in,
                              void* d_out, int out_size, void* d_ws, size_t ws_size,
                              hipStream_t stream) {
    const float* fe     = (const float*)d_in[0];
    const float* ofe    = (const float*)d_in[1];
    const float* Wm     = (const float*)d_in[2];
    const float* bias   = (const float*)d_in[3];
    const int*   bridge = (const int*)d_in[4];
    float* out = (float*)d_out;
    float* ws  = (float*)d_ws;

    float* X      = ws + OFF_X;                           // 3200 x 16 x 256
    float* Xcur   = X;
    float* Xother = X + (size_t)NROWS_ * T_ * PROJ_;
    float* A      = ws + OFF_A;
    float* numer  = ws + OFF_NUMER;
    float* sig2   = ws + OFF_SIG2;
    float* aa     = ws + OFF_AA;
    float* lossb  = ws + OFF_LOSS;
    float* htb    = ws + OFF_HT;
    int*   ints   = (int*)(ws + OFF_INTS);
    int*   counts = ints + I_COUNT;
    int*   offs   = ints + I_OFFS;
    int*   cursor = ints + I_CURSOR;
    int*   groupT = ints + I_GROUPT;
    int*   perm   = ints + I_PERM;
    float* cross  = ws + OFF_CROSS;

    k_init<<<1, 256, 0, stream>>>(ints);
    k_proj<<<dim3(MTOT_ / 128, PROJ_ / 64), 256, 0, stream>>>(fe, Wm, bias, Xcur);
    k_proj<<<dim3(MTOT_ / 128, PROJ_ / 64), 256, 0, stream>>>(ofe, Wm, bias, Xother);
    k_norm<<<NNEG_ * T_, 256, 0, stream>>>(X);
    k_bridge<<<NROWS_, 256, 0, stream>>>(Xcur, bridge, A, numer, sig2, aa, counts);
    k_buckets<<<1, 1, 0, stream>>>(counts, offs, groupT);
    k_scatter<<<(NROWS_ + 255) / 256, 256, 0, stream>>>(bridge, offs, cursor, perm);
    k_cross<<<dim3(128, NNEG_ / 64), 32, 0, stream>>>(A, X, groupT, perm, cross);
    k_topk<<<NROWS_, 256, 0, stream>>>(cross, numer, sig2, aa, lossb);
    k_headtail<<<NROWS_, 256, 0, stream>>>(Xcur, htb);
    k_final<<<1, 256, 0, stream>>>(lossb, htb, out);
}